// PersonalizedADRModel_31464930411166
// MI455X (gfx1250) — compile-verified
//
#include <hip/hip_runtime.h>
#include <hip/hip_bf16.h>

typedef __attribute__((ext_vector_type(2))) float v2f;
typedef __attribute__((ext_vector_type(8))) float v8f;

#define NODE_HID 64

// ---------------------------------------------------------------------------
// deg/dinv helpers
// ---------------------------------------------------------------------------
__global__ void init_deg_kernel(float* deg, int n) {
    int t = blockIdx.x * blockDim.x + threadIdx.x;
    if (t < n) deg[t] = 1.0f;   // self-loop contributes 1 to every node's degree
}

__global__ void deg_accum_kernel(const int* __restrict__ dst, float* deg, int e) {
    int t = blockIdx.x * blockDim.x + threadIdx.x;
    if (t < e) atomicAdd(&deg[dst[t]], 1.0f);
}

__global__ void dinv_kernel(float* deg, int n) {
    int t = blockIdx.x * blockDim.x + threadIdx.x;
    if (t < n) {
        float d = deg[t];
        deg[t] = (d > 0.0f) ? rsqrtf(d) : 0.0f;
    }
}

// ---------------------------------------------------------------------------
// Dense GEMM: D[N x 64] = A[N x K] @ B[K x 64], f32 WMMA 16x16x4.
// One wave per 16x16 tile; block = 4 waves = the 4 column tiles of a row tile.
// Requires N % 16 == 0 (holds: N = 100000).
// ---------------------------------------------------------------------------
__global__ void gemm_wmma_f32_kernel(const float* __restrict__ A,
                                     const float* __restrict__ B,
                                     float* __restrict__ D, int K) {
    const int lane = threadIdx.x & 31;
    const int wave = threadIdx.x >> 5;          // 0..3 : column tile
    const int half = lane >> 4;                  // 0: K lo / rows 0-7, 1: K hi / rows 8-15
    const int m    = lane & 15;
    const int row0 = blockIdx.x * 16;
    const int n0   = wave * 16;

    v8f acc = {};
    for (int k0 = 0; k0 < K; k0 += 4) {
        v2f a, b;
        // A fragment (16x4): lane half 0 -> K = k0,k0+1 ; half 1 -> K = k0+2,k0+3
        a.x = A[(size_t)(row0 + m) * K + k0 + (half ? 2 : 0)];
        a.y = A[(size_t)(row0 + m) * K + k0 + (half ? 3 : 1)];
        // B fragment (4x16), row-major B[K x 64]
        b.x = B[(size_t)(k0 + (half ? 2 : 0)) * NODE_HID + n0 + m];
        b.y = B[(size_t)(k0 + (half ? 3 : 1)) * NODE_HID + n0 + m];
        acc = __builtin_amdgcn_wmma_f32_16x16x4_f32(false, a, false, b,
                                                    (short)0, acc, false, false);
    }
#pragma unroll
    for (int j = 0; j < 8; ++j) {
        D[(size_t)(row0 + j + half * 8) * NODE_HID + n0 + m] = acc[j];
    }
}

// ---------------------------------------------------------------------------
// Zero an f32 buffer (float4 stores)
// ---------------------------------------------------------------------------
__global__ void zero4_kernel(float4* p, int n4) {
    int t = blockIdx.x * blockDim.x + threadIdx.x;
    if (t < n4) p[t] = make_float4(0.f, 0.f, 0.f, 0.f);
}

// ---------------------------------------------------------------------------
// Edge scatter: agg[dst] += h[src] * dinv[src]*dinv[dst].
// 4 threads per edge, 16 columns each; h rows are L2-resident.
// ---------------------------------------------------------------------------
__global__ void scatter_kernel(const int* __restrict__ src,
                               const int* __restrict__ dst,
                               const float* __restrict__ dinv,
                               const float* __restrict__ h,
                               float* agg, int e) {
    int t = blockIdx.x * blockDim.x + threadIdx.x;
    int edge = t >> 2;
    if (edge >= e) return;
    int q = t & 3;
    int s = src[edge];
    int d = dst[edge];
    float norm = dinv[s] * dinv[d];
    const float4* hp = (const float4*)(h + (size_t)s * NODE_HID + q * 16);
    float* ap = agg + (size_t)d * NODE_HID + q * 16;
#pragma unroll
    for (int i = 0; i < 4; ++i) {
        float4 v = hp[i];
        atomicAdd(ap + i * 4 + 0, v.x * norm);
        atomicAdd(ap + i * 4 + 1, v.y * norm);
        atomicAdd(ap + i * 4 + 2, v.z * norm);
        atomicAdd(ap + i * 4 + 3, v.w * norm);
    }
}

// ---------------------------------------------------------------------------
// Combine: x = (agg + h*dinv^2 + bias), optional ReLU. (Self-loop fused here.)
// ---------------------------------------------------------------------------
__global__ void combine_kernel(const float* __restrict__ agg,
                               const float* __restrict__ h,
                               const float* __restrict__ dinv,
                               const float* __restrict__ bias,
                               float* __restrict__ xout,
                               int total, int doRelu) {
    int t = blockIdx.x * blockDim.x + threadIdx.x;
    if (t >= total) return;
    int node = t >> 6;
    int c = t & 63;
    float di = dinv[node];
    float v = agg[t] + h[t] * di * di + bias[c];
    if (doRelu) v = fmaxf(v, 0.0f);
    xout[t] = v;
}

// ---------------------------------------------------------------------------
// Pair MLP: per wave, 16 pairs.
//   combined[16 x 72] = [ x[p0]*x[p1] (64) | patient_features (8) ]   (LDS)
//   h = relu(combined @ Wf1 + bf1)          -> 4 col tiles, 18 WMMA K-steps
//   out = sigmoid(h @ Wf2 + bf2)            -> shfl_xor 16-lane reduction
// ---------------------------------------------------------------------------
#define PAIR_LDS_STRIDE 73
__global__ void pair_mlp_kernel(const int* __restrict__ pairs,
                                const float* __restrict__ pf,
                                const float* __restrict__ x2,
                                const float* __restrict__ Wf1,
                                const float* __restrict__ bf1,
                                const float* __restrict__ Wf2,
                                const float* __restrict__ bf2,
                                float* __restrict__ out,
                                int P, int numTiles) {
    const int lane = threadIdx.x & 31;
    const int wave = threadIdx.x >> 5;
    const int tile = blockIdx.x * 8 + wave;
    const bool active = tile < numTiles;
    __shared__ float lds[8 * 16 * PAIR_LDS_STRIDE];
    float* tl = &lds[wave * 16 * PAIR_LDS_STRIDE];
    const int row0 = tile * 16;

    if (active) {
        for (int idx = lane; idx < 16 * 72; idx += 32) {
            int r = idx / 72;
            int k = idx - r * 72;
            int p = row0 + r;
            if (p >= P) p = P - 1;   // clamp (P % 16 == 0 in practice)
            float v;
            if (k < 64) {
                int nd = pairs[2 * p];
                int na = pairs[2 * p + 1];
                v = x2[(size_t)nd * NODE_HID + k] * x2[(size_t)na * NODE_HID + k];
            } else {
                v = pf[(size_t)p * 8 + (k - 64)];
            }
            tl[r * PAIR_LDS_STRIDE + k] = v;
        }
    }
    __syncthreads();
    if (!active) return;

    const int half = lane >> 4;
    const int m    = lane & 15;

    // Preload the A-operand fragments (16x72 -> 18 K-steps), reused by all 4
    // column tiles.
    v2f areg[18];
#pragma unroll
    for (int ks = 0; ks < 18; ++ks) {
        int k0 = ks * 4;
        areg[ks].x = tl[m * PAIR_LDS_STRIDE + k0 + (half ? 2 : 0)];
        areg[ks].y = tl[m * PAIR_LDS_STRIDE + k0 + (half ? 3 : 1)];
    }

    float sum[8];
#pragma unroll
    for (int j = 0; j < 8; ++j) sum[j] = 0.0f;

    for (int nt = 0; nt < 4; ++nt) {
        const int n0 = nt * 16;
        v8f acc = {};
#pragma unroll
        for (int ks = 0; ks < 18; ++ks) {
            int k0 = ks * 4;
            v2f b;
            b.x = Wf1[(size_t)(k0 + (half ? 2 : 0)) * 64 + n0 + m];
            b.y = Wf1[(size_t)(k0 + (half ? 3 : 1)) * 64 + n0 + m];
            acc = __builtin_amdgcn_wmma_f32_16x16x4_f32(false, areg[ks], false, b,
                                                        (short)0, acc, false, false);
        }
        // bias + ReLU + partial dot with Wf2 (column n0+m lives in this lane)
        float bias = bf1[n0 + m];
        float w2   = Wf2[n0 + m];
#pragma unroll
        for (int j = 0; j < 8; ++j) {
            float hval = acc[j] + bias;
            hval = hval > 0.0f ? hval : 0.0f;
            sum[j] += hval * w2;
        }
    }

    // Reduce across the 16 lanes of each half (rows j+8*half).
#pragma unroll
    for (int j = 0; j < 8; ++j) {
        float s = sum[j];
        s += __shfl_xor(s, 1, 32);
        s += __shfl_xor(s, 2, 32);
        s += __shfl_xor(s, 4, 32);
        s += __shfl_xor(s, 8, 32);
        sum[j] = s;
    }

    if (m == 0) {
        float b2 = bf2[0];
#pragma unroll
        for (int j = 0; j < 8; ++j) {
            int r = row0 + j + half * 8;
            if (r < P) {
                float z = sum[j] + b2;
                out[r] = 1.0f / (1.0f + expf(-z));
            }
        }
    }
}

// ---------------------------------------------------------------------------
// Launch
// ---------------------------------------------------------------------------
extern "C" void kernel_launch(void* const* d_in, const int* in_sizes, int n_in,
                              void* d_out, int out_size, void* d_ws, size_t ws_size,
                              hipStream_t stream) {
    const int* edge_index = (const int*)d_in[0];     // (2, E)
    const int* edge_pairs = (const int*)d_in[1];     // (P, 2)
    const float* pfeat    = (const float*)d_in[2];   // (P, 8)
    const float* emb      = (const float*)d_in[3];   // (N, 32)
    const float* W1       = (const float*)d_in[4];   // (32, 64)
    const float* b1       = (const float*)d_in[5];   // (64,)
    const float* W2       = (const float*)d_in[6];   // (64, 64)
    const float* b2       = (const float*)d_in[7];   // (64,)
    const float* Wf1      = (const float*)d_in[8];   // (72, 64)
    const float* bf1      = (const float*)d_in[9];   // (64,)
    const float* Wf2      = (const float*)d_in[10];  // (64, 1)
    const float* bf2      = (const float*)d_in[11];  // (1,)
    float* outp           = (float*)d_out;           // (P,)

    const int E = in_sizes[0] / 2;
    const int P = in_sizes[1] / 2;
    const int N = in_sizes[3] / 32;

    const int* srcp = edge_index;        // edge_index[0]
    const int* dstp = edge_index + E;    // edge_index[1]

    // Workspace layout (floats): dinv[N] | hbuf[N*64] | aggbuf[N*64] | xbuf[N*64]
    float* ws     = (float*)d_ws;
    float* dinv   = ws;
    float* hbuf   = dinv + N;
    float* aggbuf = hbuf + (size_t)N * NODE_HID;
    float* xbuf   = aggbuf + (size_t)N * NODE_HID;

    const int B = 256;
    const int feTotal = N * NODE_HID;
    const int feBlocks = (feTotal + B - 1) / B;
    const int zBlocks  = (feTotal / 4 + B - 1) / B;
    const int scBlocks = (E * 4 + B - 1) / B;
    const int rowTiles = N / 16;             // N = 100000 -> 6250 exact

    // ---- degrees & dinv ----
    init_deg_kernel<<<(N + B - 1) / B, B, 0, stream>>>(dinv, N);
    deg_accum_kernel<<<(E + B - 1) / B, B, 0, stream>>>(dstp, dinv, E);
    dinv_kernel<<<(N + B - 1) / B, B, 0, stream>>>(dinv, N);

    // ---- layer 1: h = emb @ W1 ; aggregate ; relu(+b1) ----
    gemm_wmma_f32_kernel<<<rowTiles, 128, 0, stream>>>(emb, W1, hbuf, 32);
    zero4_kernel<<<zBlocks, B, 0, stream>>>((float4*)aggbuf, feTotal / 4);
    scatter_kernel<<<scBlocks, B, 0, stream>>>(srcp, dstp, dinv, hbuf, aggbuf, E);
    combine_kernel<<<feBlocks, B, 0, stream>>>(aggbuf, hbuf, dinv, b1, xbuf, feTotal, 1);

    // ---- layer 2: h = x @ W2 ; aggregate ; (+b2) ----
    gemm_wmma_f32_kernel<<<rowTiles, 128, 0, stream>>>(xbuf, W2, hbuf, 64);
    zero4_kernel<<<zBlocks, B, 0, stream>>>((float4*)aggbuf, feTotal / 4);
    scatter_kernel<<<scBlocks, B, 0, stream>>>(srcp, dstp, dinv, hbuf, aggbuf, E);
    combine_kernel<<<feBlocks, B, 0, stream>>>(aggbuf, hbuf, dinv, b2, xbuf, feTotal, 0);

    // ---- pair MLP ----
    const int numTiles = (P + 15) / 16;      // 62500 exact for P = 1e6
    const int pairBlocks = (numTiles + 7) / 8;
    pair_mlp_kernel<<<pairBlocks, 256, 0, stream>>>(edge_pairs, pfeat, xbuf,
                                                    Wf1, bf1, Wf2, bf2, outp,
                                                    P, numTiles);
}